// UnbalancedSinkhornLoss_38182259261658
// MI455X (gfx1250) — compile-verified
//
#include <hip/hip_runtime.h>
#include <math.h>

// ---------------------------------------------------------------------------
// Unbalanced debiased Sinkhorn (geomloss, p=2, blur=0.1, reach=1.0) on CDNA5.
// Softmin in exp-domain == batched GEMM (B x 32) @ (32 x 32) -> FP32 WMMA
// (v_wmma_f32_16x16x4_f32 chained 8x over K=32). All state LDS-resident;
// inputs are read from global memory twice (8 MB per pass, trivial vs 23.3 TB/s).
// ---------------------------------------------------------------------------

typedef __attribute__((ext_vector_type(2))) float v2f;
typedef __attribute__((ext_vector_type(8))) float v8f;

#define NPTS     32     // points per batch (fixed by problem)
#define BB       32     // batches per block (2 waves x 16 batch-rows)
#define NTHREADS 64     // 2 wave32's
#define NELEM    (BB * NPTS)   // 1024

struct SmemT {
  float alog[NELEM];
  float blog[NELEM];
  float f_aa[NELEM];
  float g_bb[NELEM];
  float g_ab[NELEM];
  float f_ba[NELEM];
  float tmp [NELEM];
  float hbuf[NELEM];
  float cbuf[NPTS * NPTS];   // C[i][j] = 0.5*|p_i - p_j|^2
  float kbuf[NPTS * NPTS];   // exp(-C/eps), rebuilt per eps
  float mbuf[BB];            // per-batch row max of h
  float red [NTHREADS];
};

// Rebuild K_eps = exp(-C/eps) in LDS.
__device__ __forceinline__ void build_kmat(SmemT& s, float inv_eps, int tid) {
  __syncthreads();
  for (int idx = tid; idx < NPTS * NPTS; idx += NTHREADS)
    s.kbuf[idx] = __expf(-s.cbuf[idx] * inv_eps);
  __syncthreads();
}

// dst = (blend ? 0.5*(dst + d*softmin) : d*softmin), softmin over h = logw + pot/eps.
// WMMA layout (32-bit, 16x16x4):
//   A (16x4): lanes 0-15 hold M=lane, K={k0,k0+1}; lanes 16-31 hold M=lane-16, K={k0+2,k0+3}
//   B (4x16): lanes 0-15 hold N=lane, K={k0,k0+1}; lanes 16-31 hold N=lane-16, K={k0+2,k0+3}
//   D (16x16): lanes 0-15: N=lane, M=r (vgpr r); lanes 16-31: N=lane-16, M=8+r
__device__ __forceinline__ void softmin_step(
    float eps, float inv_eps, float dampf,
    const float* __restrict__ logw, const float* pot,
    float* dst, bool blend,
    SmemT& s, int tid, int lane, int wave)
{
  __syncthreads();
  // h = logw + pot/eps
  for (int idx = tid; idx < NELEM; idx += NTHREADS) {
    float h = logw[idx];
    if (pot) h += pot[idx] * inv_eps;
    s.hbuf[idx] = h;
  }
  __syncthreads();
  // per-batch stabilizer m_b = max_j h[b,j]
  if (tid < BB) {
    float m = -3.0e38f;
    #pragma unroll
    for (int j = 0; j < NPTS; ++j) m = fmaxf(m, s.hbuf[tid * NPTS + j]);
    s.mbuf[tid] = m;
  }
  __syncthreads();

  const int base = wave * 16;            // first batch row of this wave
  const int half = lane >> 4;            // 0 or 1
  const int arow = base + (lane & 15);   // A-matrix M row (batch)
  const float mv = s.mbuf[arow];

  // A operand: V[b,j] = exp(h - m_b), per-lane contiguous K pairs
  v2f av[8];
  #pragma unroll
  for (int kk = 0; kk < 8; ++kk) {
    const int koff = 4 * kk + 2 * half;
    v2f hp = *reinterpret_cast<const v2f*>(&s.hbuf[arow * NPTS + koff]);
    v2f e;
    e.x = __expf(hp.x - mv);
    e.y = __expf(hp.y - mv);
    av[kk] = e;
  }

  #pragma unroll
  for (int ntile = 0; ntile < 2; ++ntile) {
    const int n = (lane & 15) + 16 * ntile;   // output column i == Kmat row i
    v8f acc = {};
    #pragma unroll
    for (int kk = 0; kk < 8; ++kk) {
      const int koff = 4 * kk + 2 * half;
      v2f bv = *reinterpret_cast<const v2f*>(&s.kbuf[n * NPTS + koff]);
      // (neg_a, A, neg_b, B, c_mod, C, reuse_a, reuse_b)
      acc = __builtin_amdgcn_wmma_f32_16x16x4_f32(
          false, av[kk], false, bv, (short)0, acc, false, false);
    }
    #pragma unroll
    for (int r = 0; r < 8; ++r) {
      const int brow = base + r + 8 * half;            // batch row of acc[r]
      const float S  = acc[r];
      const float sm = -eps * (s.mbuf[brow] + __logf(S));
      const float val = dampf * sm;
      float* p = &dst[brow * NPTS + n];
      if (blend) *p = 0.5f * (*p + val); else *p = val;
    }
  }
}

__global__ void __launch_bounds__(NTHREADS)
sinkhorn_kernel(const float* __restrict__ pred, const float* __restrict__ target,
                const float* __restrict__ pos, float* __restrict__ partials, int B)
{
  __shared__ SmemT s;
  const int tid  = threadIdx.x;
  const int lane = tid & 31;
  const int wave = tid >> 5;
  const long long gofs = (long long)blockIdx.x * NELEM;

  // C[i][j] = 0.5 * ((z_i-z_j)^2 + (y_i-y_j)^2)
  for (int idx = tid; idx < NPTS * NPTS; idx += NTHREADS) {
    const int i = idx >> 5, j = idx & 31;
    const float dz = pos[2 * i]     - pos[2 * j];
    const float dy = pos[2 * i + 1] - pos[2 * j + 1];
    s.cbuf[idx] = 0.5f * (dz * dz + dy * dy);
  }
  // log measures (first of two global reads of the 8 MB inputs)
  for (int idx = tid; idx < NELEM; idx += NTHREADS) {
    s.alog[idx] = __logf(pred  [gofs + idx]);
    s.blog[idx] = __logf(target[gofs + idx]);
  }
  __syncthreads();

  // geomloss epsilon_schedule(p=2, diameter=2*sqrt(2), blur=0.1, scaling=0.5)
  const float eps_list[7] = {8.f, 8.f, 2.f, 0.5f, 0.125f, 0.03125f, 0.01f};

  // ---- init at eps0 = 8 (f_aa==g_ab and g_bb==f_ba initially) ----
  {
    const float eps = 8.f, inv = 0.125f, d = 1.f / 9.f;
    build_kmat(s, inv, tid);
    softmin_step(eps, inv, d, s.alog, nullptr, s.g_ab, false, s, tid, lane, wave);
    softmin_step(eps, inv, d, s.blog, nullptr, s.f_ba, false, s, tid, lane, wave);
    __syncthreads();
    for (int idx = tid; idx < NELEM; idx += NTHREADS) {
      s.f_aa[idx] = s.g_ab[idx];
      s.g_bb[idx] = s.f_ba[idx];
    }
  }

  // ---- annealed symmetric loop ----
  for (int it = 0; it < 7; ++it) {
    const float eps = eps_list[it];
    const float inv = 1.f / eps;
    const float d   = 1.f / (1.f + eps);
    build_kmat(s, inv, tid);
    // ft -> tmp (uses old g_ab); gt blended into g_ab (uses old f_ba)
    softmin_step(eps, inv, d, s.blog, s.g_ab, s.tmp,  false, s, tid, lane, wave);
    softmin_step(eps, inv, d, s.alog, s.f_ba, s.g_ab, true,  s, tid, lane, wave);
    __syncthreads();
    for (int idx = tid; idx < NELEM; idx += NTHREADS)
      s.f_ba[idx] = 0.5f * (s.f_ba[idx] + s.tmp[idx]);
    // symmetric self-updates
    softmin_step(eps, inv, d, s.alog, s.f_aa, s.f_aa, true, s, tid, lane, wave);
    softmin_step(eps, inv, d, s.blog, s.g_bb, s.g_bb, true, s, tid, lane, wave);
  }

  // ---- final extrapolation at eps = blur^2 = 0.01 ----
  {
    const float eps = 0.01f, inv = 100.f, d = 1.f / 1.01f;
    build_kmat(s, inv, tid);
    softmin_step(eps, inv, d, s.blog, s.g_ab, s.tmp,  false, s, tid, lane, wave); // new f_ba
    softmin_step(eps, inv, d, s.alog, s.f_ba, s.g_ab, false, s, tid, lane, wave); // new g_ab (old f_ba)
    __syncthreads();
    for (int idx = tid; idx < NELEM; idx += NTHREADS)
      s.f_ba[idx] = s.tmp[idx];
    softmin_step(eps, inv, d, s.alog, s.f_aa, s.f_aa, false, s, tid, lane, wave);
    softmin_step(eps, inv, d, s.blog, s.g_bb, s.g_bb, false, s, tid, lane, wave);
  }

  __syncthreads();
  // ---- unbalanced debiased cost, mean over batches ----
  const float scale = (1.0f + 0.5f * 0.01f) / (float)B;   // (rho + eps/2)/B, rho=1
  float part = 0.f;
  for (int idx = tid; idx < NELEM; idx += NTHREADS) {
    const float pa = pred  [gofs + idx];
    const float tb = target[gofs + idx];
    part += pa * (__expf(-s.f_aa[idx]) - __expf(-s.f_ba[idx]))
          + tb * (__expf(-s.g_bb[idx]) - __expf(-s.g_ab[idx]));
  }
  s.red[tid] = part * scale;
  __syncthreads();
  #pragma unroll
  for (int off = NTHREADS / 2; off > 0; off >>= 1) {
    if (tid < off) s.red[tid] += s.red[tid + off];
    __syncthreads();
  }
  if (tid == 0) partials[blockIdx.x] = s.red[0];
}

// Deterministic fixed-order final reduction (replay-safe, no atomics).
__global__ void __launch_bounds__(256)
reduce_kernel(const float* __restrict__ partials, float* __restrict__ out, int nblocks)
{
  __shared__ float red[256];
  const int tid = threadIdx.x;
  float acc = 0.f;
  for (int i = tid; i < nblocks; i += 256) acc += partials[i];
  red[tid] = acc;
  __syncthreads();
  #pragma unroll
  for (int off = 128; off > 0; off >>= 1) {
    if (tid < off) red[tid] += red[tid + off];
    __syncthreads();
  }
  if (tid == 0) out[0] = red[0];
}

extern "C" void kernel_launch(void* const* d_in, const int* in_sizes, int n_in,
                              void* d_out, int out_size, void* d_ws, size_t ws_size,
                              hipStream_t stream)
{
  const float* pred   = (const float*)d_in[0];   // (B, 32) f32
  const float* target = (const float*)d_in[1];   // (B, 32) f32
  const float* pos    = (const float*)d_in[2];   // (32, 2) f32
  const int B = in_sizes[0] / NPTS;              // 32768
  const int nblocks = B / BB;                    // 1024

  float* partials = (float*)d_ws;                // nblocks floats of scratch
  sinkhorn_kernel<<<nblocks, NTHREADS, 0, stream>>>(pred, target, pos, partials, B);
  reduce_kernel<<<1, 256, 0, stream>>>(partials, (float*)d_out, nblocks);
}